// HeteroGNN_38173669327124
// MI455X (gfx1250) — compile-verified
//
#include <hip/hip_runtime.h>
#include <math.h>

// ---------------------------------------------------------------------------
// Problem constants (from the reference)
// ---------------------------------------------------------------------------
constexpr int    NN   = 50000;             // nodes per ntype (50000 = 3125*16)
constexpr int    EE   = 500000;            // edges per relation
constexpr int    OUTF = 128;               // out feats (= in feats)
constexpr int    HH   = 8;                 // heads
constexpr int    DD   = 16;                // head dim
constexpr size_t NF   = (size_t)NN * OUTF; // floats per [N,128] matrix

// ---------------------------------------------------------------------------
// params blob offsets (floats). Assumes JAX pytree flattening = sorted dict
// keys, recursively:
//   agg:  W1[256], b1[16], b2[1], w2[16]
//   bn :  herb{b,g}, ing{b,g}, tar{b,g}  (128 each)
//   rel:  'h-i','i-h','i-t','t-i'; each: Wdst[16384], Wsrc[16384],
//         attn[128], bias[128]
// ---------------------------------------------------------------------------
constexpr int AGG_W1 = 0, AGG_B1 = 256, AGG_B2 = 272, AGG_W2 = 273;
constexpr int BN_HERB = 289, BN_ING = 545, BN_TAR = 801;  // +0 = beta, +128 = gamma
constexpr int RELBASE = 1057, RELSTRIDE = 33024;
constexpr int R_HI = RELBASE + 0 * RELSTRIDE;
constexpr int R_IH = RELBASE + 1 * RELSTRIDE;
constexpr int R_IT = RELBASE + 2 * RELSTRIDE;
constexpr int R_TI = RELBASE + 3 * RELSTRIDE;
constexpr int W_DST = 0, W_SRC = 16384, W_ATTN = 32768, W_BIAS = 32896;

typedef float v2f __attribute__((ext_vector_type(2)));
typedef float v8f __attribute__((ext_vector_type(8)));

#if defined(__has_builtin)
#if __has_builtin(__builtin_amdgcn_wmma_f32_16x16x4_f32)
#define HAVE_WMMA_F32X4 1
#endif
#endif

// ---------------------------------------------------------------------------
// GEMM: Y[M,128] = X[M,128] @ W[128,128], exact fp32 via V_WMMA_F32_16X16X4_F32
// One wave per 16x16 output tile; 8 waves per block share a 16-row strip.
// A layout (16x4 f32): lane L: M = L&15, VGPR0 = K(k0+2*(L>>4)), VGPR1 = K+1
// B layout (4x16 f32): lane L: N = L&15, VGPR0 = K(k0+2*(L>>4)), VGPR1 = K+1
// C/D  (16x16 f32)   : VGPR v: M = v + 8*(L>>4), N = L&15
// ---------------------------------------------------------------------------
#ifdef HAVE_WMMA_F32X4
__global__ __launch_bounds__(256) void gemm128_wmma(const float* __restrict__ X,
                                                    const float* __restrict__ W,
                                                    float* __restrict__ Y, int M) {
  const int wave  = (blockIdx.x * blockDim.x + threadIdx.x) >> 5;
  const int lane  = threadIdx.x & 31;
  const int row_t = wave >> 3;     // 128/16 = 8 column tiles
  const int col_t = wave & 7;
  if (row_t * 16 >= M) return;
  const int m    = lane & 15;
  const int half = lane >> 4;
  const int arow = row_t * 16 + m;   // A row for this lane
  const int bcol = col_t * 16 + m;   // B col for this lane
  v8f acc = {};
#pragma unroll 4
  for (int k0 = 0; k0 < 128; k0 += 4) {
    const int ka = k0 + 2 * half;
    v2f a, b;
    a.x = X[(size_t)arow * 128 + ka];
    a.y = X[(size_t)arow * 128 + ka + 1];
    b.x = W[(size_t)ka * 128 + bcol];
    b.y = W[(size_t)(ka + 1) * 128 + bcol];
    acc = __builtin_amdgcn_wmma_f32_16x16x4_f32(false, a, false, b,
                                                (short)0, acc, false, false);
  }
#pragma unroll
  for (int v = 0; v < 8; ++v) {
    const int rr = row_t * 16 + v + 8 * half;
    Y[(size_t)rr * 128 + col_t * 16 + (lane & 15)] = acc[v];
  }
}
#else
// Fallback (keeps fp32 exactness): one thread per output element.
__global__ __launch_bounds__(256) void gemm128_naive(const float* __restrict__ X,
                                                     const float* __restrict__ W,
                                                     float* __restrict__ Y, int M) {
  const size_t t = (size_t)blockIdx.x * blockDim.x + threadIdx.x;
  if (t >= (size_t)M * 128) return;
  const int r = (int)(t >> 7), c = (int)(t & 127);
  float acc = 0.f;
  for (int k = 0; k < 128; ++k) acc = fmaf(X[(size_t)r * 128 + k], W[(size_t)k * 128 + c], acc);
  Y[t] = acc;
}
#endif

// ---------------------------------------------------------------------------
// Small utility kernels
// ---------------------------------------------------------------------------
__global__ __launch_bounds__(256) void fill_kernel(float* __restrict__ p, float v, size_t n) {
  size_t i = (size_t)blockIdx.x * blockDim.x + threadIdx.x;
  if (i < n) p[i] = v;
}

__global__ __launch_bounds__(256) void bias_init_kernel(float* __restrict__ out,
                                                        const float* __restrict__ bias,
                                                        size_t n_elems) {
  size_t i = (size_t)blockIdx.x * blockDim.x + threadIdx.x;
  if (i < n_elems) out[i] = bias[i & 127];
}

// ---------------------------------------------------------------------------
// Edge pass A: logits[e,h] = dot(leaky_relu(fs[src]+fd[dst], 0.2), attn[h])
//              + atomic segment-max into mmax[dst,h]
// One thread per (edge, head): 2x float4x4 gathers (64B-aligned slices).
// ---------------------------------------------------------------------------
__global__ __launch_bounds__(256) void logits_max_kernel(
    const float* __restrict__ fs, const float* __restrict__ fd,
    const int* __restrict__ src, const int* __restrict__ dst,
    const float* __restrict__ attn, float* __restrict__ logits,
    float* __restrict__ mmax, int E) {
  const int t = blockIdx.x * blockDim.x + threadIdx.x;
  if (t >= E * HH) return;
  const int e = t >> 3, h = t & 7;
  const int s = src[e], d = dst[e];
  const float4* ps = (const float4*)(fs + (size_t)s * 128 + h * 16);
  const float4* pd = (const float4*)(fd + (size_t)d * 128 + h * 16);
  const float4* pa = (const float4*)(attn + h * 16);
  float acc = 0.f;
#pragma unroll
  for (int q = 0; q < 4; ++q) {
    float4 a = ps[q], b = pd[q], w = pa[q];
    float v;
    v = a.x + b.x; v = v > 0.f ? v : 0.2f * v; acc = fmaf(v, w.x, acc);
    v = a.y + b.y; v = v > 0.f ? v : 0.2f * v; acc = fmaf(v, w.y, acc);
    v = a.z + b.z; v = v > 0.f ? v : 0.2f * v; acc = fmaf(v, w.z, acc);
    v = a.w + b.w; v = v > 0.f ? v : 0.2f * v; acc = fmaf(v, w.w, acc);
  }
  logits[t] = acc;
  __hip_atomic_fetch_max(&mmax[(size_t)d * HH + h], acc,
                         __ATOMIC_RELAXED, __HIP_MEMORY_SCOPE_AGENT);
}

// Edge pass B: ex = exp(logit - m[dst]); store over logits; atomic sum -> denom
__global__ __launch_bounds__(256) void exp_sum_kernel(
    float* __restrict__ logits, const int* __restrict__ dst,
    const float* __restrict__ mmax, float* __restrict__ denom, int E) {
  const int t = blockIdx.x * blockDim.x + threadIdx.x;
  if (t >= E * HH) return;
  const int e = t >> 3, h = t & 7;
  const int d = dst[e];
  const float ex = __expf(logits[t] - mmax[(size_t)d * HH + h]);
  logits[t] = ex;
  atomicAdd(&denom[(size_t)d * HH + h], ex);
}

// Edge pass C: out[dst,h,:] += (ex/(denom[dst,h]+1e-12)) * fs[src,h,:]
__global__ __launch_bounds__(256) void scatter_kernel(
    const float* __restrict__ ex, const float* __restrict__ fs,
    const int* __restrict__ src, const int* __restrict__ dst,
    const float* __restrict__ denom, float* __restrict__ out, int E) {
  const int t = blockIdx.x * blockDim.x + threadIdx.x;
  if (t >= E * HH) return;
  const int e = t >> 3, h = t & 7;
  const int s = src[e], d = dst[e];
  const float alpha = ex[t] / (denom[(size_t)d * HH + h] + 1e-12f);
  const float4* pf = (const float4*)(fs + (size_t)s * 128 + h * 16);
  float* po = out + (size_t)d * 128 + h * 16;
#pragma unroll
  for (int q = 0; q < 4; ++q) {
    float4 a = pf[q];
    atomicAdd(po + 4 * q + 0, alpha * a.x);
    atomicAdd(po + 4 * q + 1, alpha * a.y);
    atomicAdd(po + 4 * q + 2, alpha * a.z);
    atomicAdd(po + 4 * q + 3, alpha * a.w);
  }
}

// ---------------------------------------------------------------------------
// Cross-relation attention over 2 relations (ingredient nodes):
// one thread per (node, head).
// ---------------------------------------------------------------------------
__global__ __launch_bounds__(256) void agg2_kernel(const float* __restrict__ t0buf,
                                                   const float* __restrict__ t1buf,
                                                   const float* __restrict__ P,
                                                   float* __restrict__ out) {
  const int t = blockIdx.x * blockDim.x + threadIdx.x;
  if (t >= NN * HH) return;
  const float* W1 = P + AGG_W1;
  const float* b1 = P + AGG_B1;
  const float  b2 = P[AGG_B2];
  const float* w2 = P + AGG_W2;
  const float* r0 = t0buf + (size_t)t * DD;
  const float* r1 = t1buf + (size_t)t * DD;
  float x0[DD], x1[DD];
#pragma unroll
  for (int i = 0; i < DD; ++i) { x0[i] = r0[i]; x1[i] = r1[i]; }
  float s0 = b2, s1 = b2;
  for (int j = 0; j < DD; ++j) {
    float a0 = b1[j], a1 = a0;
#pragma unroll
    for (int i = 0; i < DD; ++i) {
      const float w = W1[i * DD + j];
      a0 = fmaf(x0[i], w, a0);
      a1 = fmaf(x1[i], w, a1);
    }
    s0 = fmaf(tanhf(a0), w2[j], s0);
    s1 = fmaf(tanhf(a1), w2[j], s1);
  }
  const float mx = fmaxf(s0, s1);
  const float e0 = __expf(s0 - mx), e1 = __expf(s1 - mx);
  const float inv = 1.f / (e0 + e1);
  const float al0 = e0 * inv, al1 = e1 * inv;
  float* po = out + (size_t)t * DD;
#pragma unroll
  for (int i = 0; i < DD; ++i) po[i] = al0 * x0[i] + al1 * x1[i];
}

// ---------------------------------------------------------------------------
// BatchNorm: per (ntype, column) sum / sumsq, coalesced full-row reads.
// grid = (row_chunks, 3), block = 128 (thread == column)
// ---------------------------------------------------------------------------
__global__ __launch_bounds__(128) void bn_stats_kernel(const float* __restrict__ Y,
                                                       float* __restrict__ sums,
                                                       float* __restrict__ sumsq) {
  const int col = threadIdx.x;
  const int nt = blockIdx.y;
  const int CH = 512;
  const int r0 = blockIdx.x * CH;
  int r1 = r0 + CH; if (r1 > NN) r1 = NN;
  const float* base = Y + (size_t)nt * NF;
  float s = 0.f, s2 = 0.f;
  for (int r = r0; r < r1; ++r) {
    const float v = base[(size_t)r * 128 + col];
    s += v; s2 = fmaf(v, v, s2);
  }
  atomicAdd(&sums[nt * 128 + col], s);
  atomicAdd(&sumsq[nt * 128 + col], s2);
}

__global__ __launch_bounds__(256) void bn_apply_kernel(float* __restrict__ Y,
                                                       const float* __restrict__ sums,
                                                       const float* __restrict__ sumsq,
                                                       const float* __restrict__ P) {
  const size_t i = (size_t)blockIdx.x * blockDim.x + threadIdx.x;
  if (i >= 3 * NF) return;
  const int nt = (int)(i / NF);
  const int col = (int)(i & 127);
  const float inv_n = 1.0f / (float)NN;
  const float mean = sums[nt * 128 + col] * inv_n;
  const float var = sumsq[nt * 128 + col] * inv_n - mean * mean;
  const int bnoff = (nt == 0) ? BN_ING : (nt == 1) ? BN_TAR : BN_HERB;
  const float beta = P[bnoff + col];
  const float gamma = P[bnoff + 128 + col];
  float y = (Y[i] - mean) * rsqrtf(var + 1e-5f) * gamma + beta;
  Y[i] = y > 0.f ? y : 0.01f * y;
}

// ---------------------------------------------------------------------------
// Launch
// ---------------------------------------------------------------------------
extern "C" void kernel_launch(void* const* d_in, const int* in_sizes, int n_in,
                              void* d_out, int out_size, void* d_ws, size_t ws_size,
                              hipStream_t stream) {
  (void)in_sizes; (void)n_in; (void)out_size; (void)ws_size;
  const float* x_ing  = (const float*)d_in[0];
  const float* x_tar  = (const float*)d_in[1];
  const float* x_herb = (const float*)d_in[2];
  const float* P      = (const float*)d_in[3];
  const int* src_hi = (const int*)d_in[4],  * dst_hi = (const int*)d_in[5];
  const int* src_ti = (const int*)d_in[6],  * dst_ti = (const int*)d_in[7];
  const int* src_ih = (const int*)d_in[8],  * dst_ih = (const int*)d_in[9];
  const int* src_it = (const int*)d_in[10], * dst_it = (const int*)d_in[11];
  float* out = (float*)d_out;
  float* ws  = (float*)d_ws;

  // workspace layout (floats); total ~68.8M floats (~275 MB)
  float* fs_hi = ws + 0 * NF;  float* fd_hi = ws + 1 * NF;
  float* fs_ti = ws + 2 * NF;  float* fd_ti = ws + 3 * NF;
  float* fs_ih = ws + 4 * NF;  float* fd_ih = ws + 5 * NF;
  float* fs_it = ws + 6 * NF;  float* fd_it = ws + 7 * NF;
  float* out_hi = ws + 8 * NF; float* out_ti = ws + 9 * NF;
  float* logits = ws + 10 * NF;                      // E*H = 4,000,000
  float* mmax   = logits + (size_t)EE * HH;          // N*H
  float* denom  = mmax + (size_t)NN * HH;            // N*H
  float* bnsum  = denom + (size_t)NN * HH;           // 384
  float* bnsq   = bnsum + 3 * 128;                   // 384

  // --- 8 feature GEMMs ---
#ifdef HAVE_WMMA_F32X4
  const int gemm_blocks = 3125;  // 3125 row-tiles * 8 col-tiles / 8 waves
#define GEMM(X, Wp, Yp) gemm128_wmma<<<gemm_blocks, 256, 0, stream>>>((X), (Wp), (Yp), NN)
#else
  const int gemm_blocks = (int)((NF + 255) / 256);
#define GEMM(X, Wp, Yp) gemm128_naive<<<gemm_blocks, 256, 0, stream>>>((X), (Wp), (Yp), NN)
#endif
  GEMM(x_herb, P + R_HI + W_SRC, fs_hi);
  GEMM(x_ing,  P + R_HI + W_DST, fd_hi);
  GEMM(x_tar,  P + R_TI + W_SRC, fs_ti);
  GEMM(x_ing,  P + R_TI + W_DST, fd_ti);
  GEMM(x_ing,  P + R_IH + W_SRC, fs_ih);
  GEMM(x_herb, P + R_IH + W_DST, fd_ih);
  GEMM(x_ing,  P + R_IT + W_SRC, fs_it);
  GEMM(x_tar,  P + R_IT + W_DST, fd_it);
#undef GEMM

  const int nh = NN * HH;
  const int nhb = (nh + 255) / 256;
  const int efb = (EE * HH + 255) / 256;
  const int nfb = (int)((NF + 255) / 256);

  auto do_rel = [&](const float* fs, const float* fd, const int* src,
                    const int* dst, int rbase, float* orel) {
    fill_kernel<<<nhb, 256, 0, stream>>>(mmax, -3.0e38f, (size_t)nh);
    fill_kernel<<<nhb, 256, 0, stream>>>(denom, 0.0f, (size_t)nh);
    bias_init_kernel<<<nfb, 256, 0, stream>>>(orel, P + rbase + W_BIAS, NF);
    logits_max_kernel<<<efb, 256, 0, stream>>>(fs, fd, src, dst,
                                               P + rbase + W_ATTN, logits, mmax, EE);
    exp_sum_kernel<<<efb, 256, 0, stream>>>(logits, dst, mmax, denom, EE);
    scatter_kernel<<<efb, 256, 0, stream>>>(logits, fs, src, dst, denom, orel, EE);
  };

  do_rel(fs_hi, fd_hi, src_hi, dst_hi, R_HI, out_hi);        // herb -> ing
  do_rel(fs_ti, fd_ti, src_ti, dst_ti, R_TI, out_ti);        // tar  -> ing
  do_rel(fs_ih, fd_ih, src_ih, dst_ih, R_IH, out + 2 * NF);  // ing  -> herb
  do_rel(fs_it, fd_it, src_it, dst_it, R_IT, out + 1 * NF);  // ing  -> tar

  // ingredient: cross-relation attention over {hi, ti} -> out region 0
  agg2_kernel<<<nhb, 256, 0, stream>>>(out_hi, out_ti, P, out);

  // BatchNorm + leaky_relu(0.01) per ntype, in place on d_out
  fill_kernel<<<3, 256, 0, stream>>>(bnsum, 0.0f, 768);  // zeroes bnsum+bnsq
  bn_stats_kernel<<<dim3((NN + 511) / 512, 3), 128, 0, stream>>>(out, bnsum, bnsq);
  bn_apply_kernel<<<(int)((3 * NF + 255) / 256), 256, 0, stream>>>(out, bnsum, bnsq, P);
}